// FCDecoder_85804856639827
// MI455X (gfx1250) — compile-verified
//
#include <hip/hip_runtime.h>

typedef __attribute__((ext_vector_type(2))) float v2f;
typedef __attribute__((ext_vector_type(8))) float v8f;

#define NPTS  2048
#define FEAT  512
#define NBATCH 32

// One block per position p. 4 waves (128 threads, wave32).
// Layer 1: Y1[32x32] = G[32x512] x W1[p]^T  via V_WMMA_F32_16X16X4_F32 (128/wave)
// Layer 2: Y2[32x8]  = Y1 x W2[p]^T         via WMMA (8 steps, N zero-padded)
// Layer 3: out[32x3] = Y2 x W3[p]^T         via VALU
__global__ __launch_bounds__(128)
void FCDecoder_gfx1250_kernel(const float* __restrict__ glf,
                              const float* __restrict__ W1, const float* __restrict__ b1,
                              const float* __restrict__ W2, const float* __restrict__ b2,
                              const float* __restrict__ W3, const float* __restrict__ b3,
                              float* __restrict__ out)
{
    __shared__ float Y1s[32 * 33];   // [batch][chan] padded -> conflict-free strided reads
    __shared__ float Y2s[32 * 9];    // [batch][chan] padded

    const int p    = blockIdx.x;
    const int tid  = threadIdx.x;
    const int wave = tid >> 5;
    const int lane = tid & 31;
    const int l16  = lane & 15;
    const int half = lane >> 4;

    // ---------------- Layer 1 ----------------
    // Wave w -> output tile (m_tile = w&1 over batch, n_tile = w>>1 over channels)
    const int m_tile = wave & 1;
    const int n_tile = wave >> 1;
    const int m = m_tile * 16 + l16;                 // batch row   (A-matrix M)
    const int n = n_tile * 16 + l16;                 // out channel (B-matrix N)

    // A 16x4 layout: lane l (half=0) holds K=kk,kk+1 of row m; lane l+16 holds K=kk+2,kk+3.
    // B 4x16 layout: same K split per column n. Both are contiguous 8B per lane,
    // lane pairs (l, l+16) cover contiguous 16B -> clean global_load_b64 coalescing.
    const float* __restrict__ Arow = glf + (size_t)m * FEAT + 2 * half;
    const float* __restrict__ Brow = W1 + ((size_t)p * NBATCH + n) * FEAT + 2 * half;

    v8f acc = {};
    #pragma unroll 1
    for (int kc = 0; kc < FEAT; kc += 64) {
        // speculative prefetch of next streamed W1 chunk (global_prefetch_b8)
        __builtin_prefetch(Brow + kc + 64, 0, 0);
        #pragma unroll
        for (int kk = 0; kk < 64; kk += 4) {
            v2f a = *(const v2f*)(Arow + kc + kk);                               // hot in WGP$
            v2f b = __builtin_nontemporal_load((const v2f*)(Brow + kc + kk));    // stream once
            acc = __builtin_amdgcn_wmma_f32_16x16x4_f32(
                      false, a, false, b, (short)0, acc, false, false);
        }
    }

    // D layout: lane l16 -> N=l16; VGPR r -> M = r + 8*half. Add bias (per channel) and stage.
    const float bias1 = b1[(size_t)p * NBATCH + n];
    #pragma unroll
    for (int r = 0; r < 8; ++r)
        Y1s[(m_tile * 16 + r + 8 * half) * 33 + n] = acc[r] + bias1;

    __syncthreads();

    // ---------------- Layer 2 ----------------
    // All 4 waves compute (EXEC must be all-ones for WMMA); waves 0,1 commit results.
    const int  m2base = (wave & 1) * 16;
    const bool nvalid = (l16 < 8);
    const float* __restrict__ B2row =
        W2 + ((size_t)p * 8 + (nvalid ? l16 : 0)) * 32 + 2 * half;

    v8f acc2 = {};
    #pragma unroll
    for (int kk = 0; kk < 32; kk += 4) {
        const int ka = kk + 2 * half;
        v2f a2;
        a2.x = Y1s[(m2base + l16) * 33 + ka];
        a2.y = Y1s[(m2base + l16) * 33 + ka + 1];
        v2f b2v = {0.0f, 0.0f};
        if (nvalid) b2v = *(const v2f*)(B2row + kk);   // zero-pad N >= 8
        acc2 = __builtin_amdgcn_wmma_f32_16x16x4_f32(
                   false, a2, false, b2v, (short)0, acc2, false, false);
    }

    if (wave < 2 && nvalid) {
        const float bias2 = b2[(size_t)p * 8 + l16];
        #pragma unroll
        for (int r = 0; r < 8; ++r)
            Y2s[(m2base + r + 8 * half) * 9 + l16] = acc2[r] + bias2;
    }

    __syncthreads();

    // ---------------- Layer 3 ----------------
    if (tid < NBATCH * 3) {
        const int bb = tid / 3;
        const int c  = tid % 3;
        const float* __restrict__ w3 = W3 + ((size_t)p * 3 + c) * 8;
        float s = b3[(size_t)p * 3 + c];
        #pragma unroll
        for (int j = 0; j < 8; ++j)
            s += Y2s[bb * 9 + j] * w3[j];
        out[((size_t)bb * 3 + c) * NPTS + p] = s;     // [B, 3, P]
    }
}

extern "C" void kernel_launch(void* const* d_in, const int* in_sizes, int n_in,
                              void* d_out, int out_size, void* d_ws, size_t ws_size,
                              hipStream_t stream) {
    (void)in_sizes; (void)n_in; (void)out_size; (void)d_ws; (void)ws_size;
    const float* glf = (const float*)d_in[0];
    const float* W1  = (const float*)d_in[1];
    const float* b1  = (const float*)d_in[2];
    const float* W2  = (const float*)d_in[3];
    const float* b2  = (const float*)d_in[4];
    const float* W3  = (const float*)d_in[5];
    const float* b3  = (const float*)d_in[6];
    float* out = (float*)d_out;

    FCDecoder_gfx1250_kernel<<<NPTS, 128, 0, stream>>>(glf, W1, b1, W2, b2, W3, b3, out);
}